// net_28252294873826
// MI455X (gfx1250) — compile-verified
//
#include <hip/hip_runtime.h>
#include <hip/hip_bf16.h>

typedef __attribute__((ext_vector_type(2))) float v2f;
typedef __attribute__((ext_vector_type(8))) float v8f;

#define NFEAT 64
#define NCOLS 256   // 2 * H * F
#define NHEAD 2

// ---- ordered-int mapping for float atomicMax (handles negatives) ----
__device__ __forceinline__ unsigned int f2o(float f) {
  unsigned int u = __float_as_uint(f);
  return (u & 0x80000000u) ? ~u : (u | 0x80000000u);
}
__device__ __forceinline__ float o2f(unsigned int o) {
  return (o & 0x80000000u) ? __uint_as_float(o ^ 0x80000000u)
                           : __uint_as_float(~o);
}

// ---- kernel 0: init segment-max sentinels (-inf ordered) and denom=0 ----
__global__ void init_stats(unsigned int* __restrict__ mbits,
                           float* __restrict__ denom, int n) {
  int i = blockIdx.x * blockDim.x + threadIdx.x;
  if (i < n) {
    mbits[i] = 0x007FFFFFu;  // f2o(-inf)
    denom[i] = 0.0f;
  }
}

// ---- kernel 1: qk = x @ W + b via V_WMMA_F32_16X16X4_F32 -----------------
// One wave32 per 16x16 output tile. K loop: 16 steps of K=4.
// A 16x4 f32 layout: lanes 0-15 -> M rows; vgpr{0,1} = K {kk, kk+1},
// kk = 2*(lane>=16). B 4x16 symmetric. C/D: vgpr j -> M = j + 8*(lane>=16),
// N = lane&15.
__global__ void qk_gemm_wmma(const float* __restrict__ x,
                             const float* __restrict__ W,
                             const float* __restrict__ bias,
                             float* __restrict__ qk, int N) {
  const int lane  = threadIdx.x;          // 0..31
  const int l16   = lane & 15;
  const int hiHalf = lane >> 4;           // 0 or 1
  const int kk    = hiHalf << 1;          // 0 or 2
  const int m0    = blockIdx.x * 16;
  const int n0    = blockIdx.y * 16;

  int mr = m0 + l16;                      // A row this lane feeds
  if (mr >= N) mr = N - 1;                // clamp (EXEC must stay full)
  const float* xrow = x + (size_t)mr * NFEAT;
  const int    ncol = n0 + l16;           // B column this lane feeds

  v8f acc = {};
#pragma unroll
  for (int k0 = 0; k0 < NFEAT; k0 += 4) {
    v2f a, b;
    a.x = xrow[k0 + kk];
    a.y = xrow[k0 + kk + 1];
    b.x = W[(size_t)(k0 + kk)     * NCOLS + ncol];
    b.y = W[(size_t)(k0 + kk + 1) * NCOLS + ncol];
    acc = __builtin_amdgcn_wmma_f32_16x16x4_f32(
        /*neg_a=*/false, a, /*neg_b=*/false, b,
        /*c_mod=*/(short)0, acc, /*reuse_a=*/false, /*reuse_b=*/false);
  }

  const float bv = bias[ncol];
  const int mbase = m0 + hiHalf * 8;
#pragma unroll
  for (int j = 0; j < 8; ++j) {
    const int r = mbase + j;
    if (r < N) qk[(size_t)r * NCOLS + ncol] = acc[j] + bv;
  }
}

// ---- kernel 2: per-edge scores s[e,h] = q[row,h] . k[col,h]; seg-max -----
// 16 lanes per edge; each lane dots one float4 (4 feats) per head, then a
// wave32 xor-shuffle reduction within the 16-lane half.
__global__ void edge_scores(const float* __restrict__ qk,
                            const int* __restrict__ row,
                            const int* __restrict__ col,
                            float* __restrict__ s,          // [E*2]
                            unsigned int* __restrict__ mbits,
                            int E) {
  const int tid = blockIdx.x * blockDim.x + threadIdx.x;
  const int e   = tid >> 4;
  const int l16 = threadIdx.x & 15;
  if (e >= E) return;

  const int r = row[e];
  const int c = col[e];
  const float4* qb = (const float4*)(qk + (size_t)r * NCOLS);
  const float4* kb = (const float4*)(qk + (size_t)c * NCOLS);

  float dot[NHEAD];
#pragma unroll
  for (int h = 0; h < NHEAD; ++h) {
    // q at float offset h*128 -> float4 offset h*32; k at +64 -> +16
    float4 qv = qb[h * 32 + l16];
    float4 kv = kb[h * 32 + 16 + l16];
    dot[h] = qv.x * kv.x + qv.y * kv.y + qv.z * kv.z + qv.w * kv.w;
  }
#pragma unroll
  for (int off = 8; off >= 1; off >>= 1) {
#pragma unroll
    for (int h = 0; h < NHEAD; ++h) dot[h] += __shfl_xor(dot[h], off, 32);
  }
  if (l16 == 0) {
#pragma unroll
    for (int h = 0; h < NHEAD; ++h) {
      s[(size_t)e * NHEAD + h] = dot[h];
      atomicMax(&mbits[(size_t)r * NHEAD + h], f2o(dot[h]));
    }
  }
}

// ---- kernel 3: ex = exp(s - m[row]); denom += ex (in-place on s) ---------
__global__ void edge_exp(const int* __restrict__ row,
                         float* __restrict__ s,             // [E*2], in/out
                         const unsigned int* __restrict__ mbits,
                         float* __restrict__ denom,
                         int E) {
  const int idx = blockIdx.x * blockDim.x + threadIdx.x;
  if (idx >= E * NHEAD) return;
  const int e = idx >> 1;
  const int h = idx & 1;
  const int r = row[e];
  const float mv = o2f(mbits[(size_t)r * NHEAD + h]);
  const float ex = expf(s[idx] - mv);
  s[idx] = ex;
  atomicAdd(&denom[(size_t)r * NHEAD + h], ex);
}

// ---- kernel 4: p = ex / denom[row]; out = mean over heads ----------------
__global__ void finalize(const int* __restrict__ row,
                         const float* __restrict__ ex,      // [E*2]
                         const float* __restrict__ denom,
                         float* __restrict__ out, int E) {
  const int e = blockIdx.x * blockDim.x + threadIdx.x;
  if (e >= E) return;
  const int r = row[e];
  const float p0 = ex[(size_t)e * 2 + 0] / denom[(size_t)r * 2 + 0];
  const float p1 = ex[(size_t)e * 2 + 1] / denom[(size_t)r * 2 + 1];
  out[e] = 0.5f * (p0 + p1);
}

extern "C" void kernel_launch(void* const* d_in, const int* in_sizes, int n_in,
                              void* d_out, int out_size, void* d_ws, size_t ws_size,
                              hipStream_t stream) {
  const float* x  = (const float*)d_in[0];
  const float* W  = (const float*)d_in[1];
  const float* b  = (const float*)d_in[2];
  const int*   ei = (const int*)d_in[3];

  const int N = in_sizes[0] / NFEAT;  // 50000
  const int E = in_sizes[3] / 2;      // 800000
  const int* row = ei;
  const int* col = ei + E;

  // workspace layout
  float*        qk    = (float*)d_ws;                       // N*256 f32
  float*        s     = qk + (size_t)N * NCOLS;             // E*2   f32
  unsigned int* mbits = (unsigned int*)(s + (size_t)E * NHEAD); // N*2 u32
  float*        denom = (float*)(mbits + (size_t)N * NHEAD);    // N*2 f32
  float*        out   = (float*)d_out;

  // 0) init per-node stats
  const int nStats = N * NHEAD;
  init_stats<<<(nStats + 255) / 256, 256, 0, stream>>>(mbits, denom, nStats);

  // 1) qk = x @ W + b  (one wave per 16x16 tile; N = 3125*16 exactly)
  dim3 gGemm((N + 15) / 16, NCOLS / 16);
  qk_gemm_wmma<<<gGemm, dim3(32), 0, stream>>>(x, W, b, qk, N);

  // 2) per-edge scores + segment max
  {
    const long long t = (long long)E * 16;
    edge_scores<<<(unsigned)((t + 255) / 256), 256, 0, stream>>>(
        qk, row, col, s, mbits, E);
  }

  // 3) exp + segment sum
  {
    const int t = E * NHEAD;
    edge_exp<<<(t + 255) / 256, 256, 0, stream>>>(row, s, mbits, denom, E);
  }

  // 4) normalize + head mean
  finalize<<<(E + 255) / 256, 256, 0, stream>>>(row, s, denom, out, E);
}